// SimpleGCN_2370821947614
// MI455X (gfx1250) — compile-verified
//
#include <hip/hip_runtime.h>
#include <cmath>

#define FDIM 64

typedef __attribute__((ext_vector_type(2))) float v2f;
typedef __attribute__((ext_vector_type(8))) float v8f;

// ---------------------------------------------------------------------------
// Degree / normalization kernels
// ---------------------------------------------------------------------------
__global__ __launch_bounds__(256) void k_fill1(float* __restrict__ p, int n) {
    int i = blockIdx.x * blockDim.x + threadIdx.x;
    if (i < n) p[i] = 1.0f;  // self-loop contributes 1 to every node's degree
}

__global__ __launch_bounds__(256) void k_deg_edges(const long long* __restrict__ dst,
                                                   float* __restrict__ deg, int e) {
    int i = blockIdx.x * blockDim.x + threadIdx.x;
    if (i < e) {
        int d = (int)dst[i];
        __hip_atomic_fetch_add(&deg[d], 1.0f, __ATOMIC_RELAXED, __HIP_MEMORY_SCOPE_AGENT);
    }
}

__global__ __launch_bounds__(256) void k_dinv(float* __restrict__ p, int n) {
    int i = blockIdx.x * blockDim.x + threadIdx.x;
    if (i < n) p[i] = rsqrtf(p[i]);  // deg >= 1 guaranteed (self loops)
}

// ---------------------------------------------------------------------------
// H = A[rows x 64] @ W[64 x 64] using V_WMMA_F32_16X16X4_F32 (full fp32).
// One block = 128 threads = 4 waves; block computes a 16x64 output tile,
// each wave owns one 16x16 N-slab. A-tile (16x64) and W (64x64) staged in LDS.
// ---------------------------------------------------------------------------
__global__ __launch_bounds__(128) void k_gemm_wmma(const float* __restrict__ A,
                                                   const float* __restrict__ W,
                                                   float* __restrict__ H, int n_rows) {
    __shared__ float sA[16 * FDIM];     // 4 KB
    __shared__ float sB[FDIM * FDIM];   // 16 KB

    const int tid  = threadIdx.x;
    const int row0 = blockIdx.x << 4;

    // Cooperative LDS fill (A tile is 16 contiguous rows => flat 1024-float copy).
    const float* Ablk = A + (size_t)row0 * FDIM;
    for (int idx = tid * 4; idx < 16 * FDIM; idx += 128 * 4) {
        int r = row0 + (idx >> 6);
        float4 v = make_float4(0.f, 0.f, 0.f, 0.f);
        if (r < n_rows) v = *reinterpret_cast<const float4*>(Ablk + idx);
        *reinterpret_cast<float4*>(sA + idx) = v;
    }
    for (int idx = tid * 4; idx < FDIM * FDIM; idx += 128 * 4) {
        *reinterpret_cast<float4*>(sB + idx) = *reinterpret_cast<const float4*>(W + idx);
    }
    __syncthreads();

    const int lane = tid & 31;
    const int wave = tid >> 5;
    const int m    = lane & 15;   // row (A) / col (B) within tile
    const int hi   = lane >> 4;   // upper half-wave => K+2 (A/B), M+8 (C/D)
    const int n0   = wave << 4;   // this wave's N-slab

    v8f acc = {0.f, 0.f, 0.f, 0.f, 0.f, 0.f, 0.f, 0.f};

    #pragma unroll
    for (int kk = 0; kk < FDIM; kk += 4) {
        const int ka = kk + (hi << 1);
        v2f a, b;
        // A 16x4: lanes0-15 VGPR{0,1} = K{0,1}; lanes16-31 = K{2,3}
        a[0] = sA[m * FDIM + ka];
        a[1] = sA[m * FDIM + ka + 1];
        // B 4x16: same striping, lane index = N column
        b[0] = sB[ka * FDIM + n0 + m];
        b[1] = sB[(ka + 1) * FDIM + n0 + m];
        acc = __builtin_amdgcn_wmma_f32_16x16x4_f32(false, a, false, b,
                                                    (short)0, acc, false, false);
    }

    // C/D 16x16 f32: VGPR v -> M = v + 8*hi, N = n0 + m
    float* Hblk = H + (size_t)row0 * FDIM + n0 + m;
    #pragma unroll
    for (int v = 0; v < 8; ++v) {
        int r = row0 + v + (hi << 3);
        if (r < n_rows) Hblk[(size_t)(v + (hi << 3)) * FDIM] = acc[v];
    }
}

// ---------------------------------------------------------------------------
// Zero the aggregation buffer (float4 granularity).
// ---------------------------------------------------------------------------
__global__ __launch_bounds__(256) void k_zero4(float4* __restrict__ p, int n4) {
    int i = blockIdx.x * blockDim.x + threadIdx.x;
    if (i < n4) p[i] = make_float4(0.f, 0.f, 0.f, 0.f);
}

// ---------------------------------------------------------------------------
// Edge scatter: 16 threads per edge, each handles a float4 feature chunk.
// agg[dst] += h[src] * dinv[src]*dinv[dst]  (native f32 global atomics; the
// whole h/agg working set is L2-resident on MI455X's 192MB L2).
// ---------------------------------------------------------------------------
__global__ __launch_bounds__(256) void k_scatter(const long long* __restrict__ src,
                                                 const long long* __restrict__ dst,
                                                 const float* __restrict__ dinv,
                                                 const float* __restrict__ h,
                                                 float* __restrict__ agg, int e) {
    long long idx = (long long)blockIdx.x * blockDim.x + threadIdx.x;
    long long tot = (long long)e << 4;
    if (idx >= tot) return;
    int ed = (int)(idx >> 4);
    int c  = (int)(idx & 15) << 2;
    int s  = (int)src[ed];
    int d  = (int)dst[ed];
    float nrm = dinv[s] * dinv[d];
    const float4 hv = *reinterpret_cast<const float4*>(h + (size_t)s * FDIM + c);
    float* o = agg + (size_t)d * FDIM + c;
    __hip_atomic_fetch_add(o + 0, hv.x * nrm, __ATOMIC_RELAXED, __HIP_MEMORY_SCOPE_AGENT);
    __hip_atomic_fetch_add(o + 1, hv.y * nrm, __ATOMIC_RELAXED, __HIP_MEMORY_SCOPE_AGENT);
    __hip_atomic_fetch_add(o + 2, hv.z * nrm, __ATOMIC_RELAXED, __HIP_MEMORY_SCOPE_AGENT);
    __hip_atomic_fetch_add(o + 3, hv.w * nrm, __ATOMIC_RELAXED, __HIP_MEMORY_SCOPE_AGENT);
}

// ---------------------------------------------------------------------------
// Finalize in place: agg = act(agg + h*dinv^2 (self loop) + bias)
// ---------------------------------------------------------------------------
__global__ __launch_bounds__(256) void k_finalize(float* __restrict__ agg,
                                                  const float* __restrict__ h,
                                                  const float* __restrict__ dinv,
                                                  const float* __restrict__ bias,
                                                  int n, int act) {
    int idx = blockIdx.x * blockDim.x + threadIdx.x;
    if (idx >= n * FDIM) return;
    int i = idx >> 6;
    int f = idx & 63;
    float di = dinv[i];
    float v  = agg[idx] + h[idx] * (di * di) + bias[f];
    if (act) v = (v >= 0.f) ? v : 0.01f * v;   // leaky_relu(0.01)
    agg[idx] = v;
}

// ---------------------------------------------------------------------------
extern "C" void kernel_launch(void* const* d_in, const int* in_sizes, int n_in,
                              void* d_out, int out_size, void* d_ws, size_t ws_size,
                              hipStream_t stream) {
    const float*     x  = (const float*)d_in[0];
    const long long* ei = (const long long*)d_in[1];   // int64 edge_index [2, E]
    const float* W1 = (const float*)d_in[2];
    const float* b1 = (const float*)d_in[3];
    const float* W2 = (const float*)d_in[4];
    const float* b2 = (const float*)d_in[5];
    const float* W3 = (const float*)d_in[6];
    const float* b3 = (const float*)d_in[7];
    float* out = (float*)d_out;

    const int N = in_sizes[0] / FDIM;   // 100000
    const int E = in_sizes[1] / 2;      // 1600000
    const long long* src = ei;          // edge_index[0]
    const long long* dst = ei + E;      // edge_index[1]

    // Workspace: dinv[N] + h[N*64]  (~26 MB). d_out doubles as the aggregation
    // buffer every layer (stream-ordered reuse).
    float* dinv = (float*)d_ws;
    float* h    = dinv + N;

    // --- normalization: deg -> D^{-1/2} ---
    k_fill1    <<<(N + 255) / 256, 256, 0, stream>>>(dinv, N);
    k_deg_edges<<<(E + 255) / 256, 256, 0, stream>>>(dst, dinv, E);
    k_dinv     <<<(N + 255) / 256, 256, 0, stream>>>(dinv, N);

    const int nblk_gemm = (N + 15) / 16;                     // 6250
    const long long scat_items = (long long)E << 4;          // 25.6M
    const int scat_blocks = (int)((scat_items + 255) / 256);
    const int nf  = N * FDIM;
    const int nf4 = nf / 4;

    const float* Ws[3] = {W1, W2, W3};
    const float* bs[3] = {b1, b2, b3};
    const float* in = x;
    for (int l = 0; l < 3; ++l) {
        k_gemm_wmma<<<nblk_gemm, 128, 0, stream>>>(in, Ws[l], h, N);
        k_zero4    <<<(nf4 + 255) / 256, 256, 0, stream>>>((float4*)out, nf4);
        k_scatter  <<<scat_blocks, 256, 0, stream>>>(src, dst, dinv, h, out, E);
        k_finalize <<<(nf + 255) / 256, 256, 0, stream>>>(out, h, dinv, bs[l], N,
                                                          (l < 2) ? 1 : 0);
        in = out;
    }
}